// Cache_68135361184561
// MI455X (gfx1250) — compile-verified
//
#include <hip/hip_runtime.h>
#include <hip/hip_bf16.h>

// CDNA5 / gfx1250: wave32, WMMA (not MFMA).
// att[b,n] = max_{i,j} dot(q[b,i,:], k[b,n,j,:]) via V_WMMA_F32_16X16X4_F32 (exact f32),
// then per-b top-8 indices. `values` input is dead code in the reference — never read.

typedef float v2f __attribute__((ext_vector_type(2)));
typedef float v8f __attribute__((ext_vector_type(8)));

static constexpr int LSTEPS = 64;    // L
static constexpr int HID    = 1024;  // h
static constexpr int BSZ    = 8;
static constexpr int NKEYS  = 128;
static constexpr int BK     = 64;    // K-chunk staged in LDS
static constexpr int SQ     = 68;    // padded LDS row stride (floats): 68*4 % 16 == 0, bank-conflict-free frags
static constexpr int TOPK   = 8;

__global__ __launch_bounds__(256)
void att_max_kernel(const float* __restrict__ query,
                    const float* __restrict__ keys,
                    float* __restrict__ att_out)
{
    __shared__ float lds_q[LSTEPS * SQ];
    __shared__ float lds_k[LSTEPS * SQ];
    __shared__ float wave_max[8];

    const int n = blockIdx.x & (NKEYS - 1);
    const int b = blockIdx.x >> 7;

    // q[b,i,h] = query[i*8192 + b*1024 + h]
    const float* qbase = query + b * HID;
    // k[n,b,j,h] contiguous 64x1024 block
    const float* kbase = keys + (size_t)n * (BSZ * LSTEPS * HID) + (size_t)b * (LSTEPS * HID);

    const int tid  = threadIdx.x;
    const int wave = tid >> 5;
    const int lane = tid & 31;
    const int half = lane >> 4;   // K sub-slot per 16x16x4 f32 layout
    const int lrow = lane & 15;   // row (A) / col (B) index within tile

    const int ti  = wave & 3;           // i-tile 0..3
    const int tj0 = (wave >> 2) * 2;    // two j-tiles per wave
    const int tj1 = tj0 + 1;

    v8f acc0 = {};
    v8f acc1 = {};

    for (int k0 = 0; k0 < HID; k0 += BK) {
        // Cooperative stage: 64 rows x 64 floats for Q and K (4 float4 per thread each)
        #pragma unroll
        for (int r = 0; r < 4; ++r) {
            const int idx = tid + r * 256;        // 0..1023
            const int row = idx >> 4;             // 0..63
            const int c4  = (idx & 15) << 2;      // float col 0..60 step 4
            const float4 qv = *(const float4*)(qbase + (size_t)row * (BSZ * HID) + k0 + c4);
            *(float4*)&lds_q[row * SQ + c4] = qv;
            const float4 kv = *(const float4*)(kbase + (size_t)row * HID + k0 + c4);
            *(float4*)&lds_k[row * SQ + c4] = kv;
        }
        __syncthreads();

        const float* arow  = &lds_q[(ti  * 16 + lrow) * SQ];
        const float* brow0 = &lds_k[(tj0 * 16 + lrow) * SQ];
        const float* brow1 = &lds_k[(tj1 * 16 + lrow) * SQ];

        #pragma unroll
        for (int kk = 0; kk < BK; kk += 4) {
            const int c = kk + 2 * half;          // lane's K pair per ISA 16x4 f32 A/B layout
            const v2f a  = *(const v2f*)(arow  + c);
            const v2f b0 = *(const v2f*)(brow0 + c);
            const v2f b1 = *(const v2f*)(brow1 + c);
            acc0 = __builtin_amdgcn_wmma_f32_16x16x4_f32(
                       false, a, false, b0, (short)0, acc0, false, false);
            acc1 = __builtin_amdgcn_wmma_f32_16x16x4_f32(
                       false, a, false, b1, (short)0, acc1, false, false);
        }
        __syncthreads();
    }

    // Max over both 16x16 tiles held by this wave (D layout irrelevant for a full max)
    float m = acc0[0];
    #pragma unroll
    for (int i = 1; i < 8; ++i) m = fmaxf(m, acc0[i]);
    #pragma unroll
    for (int i = 0; i < 8; ++i) m = fmaxf(m, acc1[i]);
    #pragma unroll
    for (int off = 16; off > 0; off >>= 1)
        m = fmaxf(m, __shfl_xor(m, off, 32));     // wave32 reduce

    if (lane == 0) wave_max[wave] = m;
    __syncthreads();
    if (tid == 0) {
        float wm = wave_max[0];
        #pragma unroll
        for (int w = 1; w < 8; ++w) wm = fmaxf(wm, wave_max[w]);
        att_out[b * NKEYS + n] = wm;
    }
}

// Per-b top-8 over 128 scores; strict '>' keeps first-occurrence tie order like jax.lax.top_k.
// Output layout matches reference: topk_idx.T -> idx_out[kk*BSZ + b].
__global__ void topk_kernel(const float* __restrict__ att, int* __restrict__ idx_out)
{
    const int b = threadIdx.x;
    if (b >= BSZ) return;
    const float* row = att + b * NKEYS;
    int chosen[TOPK];
    for (int kk = 0; kk < TOPK; ++kk) {
        float best = -3.402823466e38f;
        int bi = 0;
        for (int j = 0; j < NKEYS; ++j) {
            bool taken = false;
            for (int t = 0; t < kk; ++t) taken |= (chosen[t] == j);
            const float v = row[j];
            if (!taken && v > best) { best = v; bi = j; }
        }
        chosen[kk] = bi;
        idx_out[kk * BSZ + b] = bi;
    }
}

extern "C" void kernel_launch(void* const* d_in, const int* in_sizes, int n_in,
                              void* d_out, int out_size, void* d_ws, size_t ws_size,
                              hipStream_t stream) {
    const float* query = (const float*)d_in[0];  // [1,64,8,1024] f32
    const float* keys  = (const float*)d_in[1];  // [128,8,65536] f32
    // d_in[2] (values) is unused by the reference's max-pooling branch: do not read it.
    (void)in_sizes; (void)n_in; (void)d_ws; (void)ws_size; (void)out_size;

    float* att_out = (float*)d_out;                    // 1024 floats [b,1,n]
    int*   idx_out = (int*)d_out + BSZ * NKEYS;        // 64 int32 [k,b]

    att_max_kernel<<<dim3(BSZ * NKEYS), dim3(256), 0, stream>>>(query, keys, att_out);
    topk_kernel<<<dim3(1), dim3(32), 0, stream>>>(att_out, idx_out);
}